// Feature_Dict_Singel_Encoder_6365141533099
// MI455X (gfx1250) — compile-verified
//
#include <hip/hip_runtime.h>
#include <math.h>

#define FEATURE_DIM 128
#define DATA_SIZE   32768
#define BATCH       16
#define INV_T       (1.0f / 0.07f)

typedef float v2f __attribute__((ext_vector_type(2)));
typedef float v8f __attribute__((ext_vector_type(8)));

// ---------------------------------------------------------------------------
// Kernel 1: six batched dot-product families via V_WMMA_F32_16X16X4_F32.
// One wave handles one (batch b, 16-row k-tile). A = gathered 16x4 f32 chunk
// (per-lane 8B load from loop-invariant row base + immediate offset).
// B columns: 0 = fea_f, 1 = fea_fenzi, 2 = fea_fenmu (shared by all 3 banks):
//   accZ (bank fenzi) : col0 = f_fenzi      col2 = fenmu_fenzi
//   accM (bank fenmu) : col0 = f_fenmu      col1 = fenzi_fenmu
//   accF (bank fringe): col1 = fenzi_f      col2 = fenmu_f
// ---------------------------------------------------------------------------
__global__ __launch_bounds__(256)
void dots_wmma_kernel(const float* __restrict__ fea_f,
                      const float* __restrict__ fea_fenzi,
                      const float* __restrict__ fea_fenmu,
                      const int*   __restrict__ idx,
                      const float* __restrict__ mem_fringe,
                      const float* __restrict__ mem_fenzi,
                      const float* __restrict__ mem_fenmu,
                      float* __restrict__ out)
{
    __shared__ float sout[8][96];                     // 8 waves * 6 outputs * 16 k

    const int lane   = threadIdx.x & 31;
    const int w      = threadIdx.x >> 5;
    const int waveId = blockIdx.x * (blockDim.x >> 5) + w;
    const int b      = waveId >> 11;                  // 2048 k-tiles per batch
    const int k0     = (waveId & 2047) << 4;          // 16 k per tile
    const int n      = lane & 15;                     // WMMA N / M-lane index
    const int half   = lane >> 4;                     // K-half selector

    // Gather row index for this lane's M-row (lanes 16-31 mirror 0-15).
    const int row = idx[b * DATA_SIZE + k0 + n];

    // A-matrix per-lane bases (16x4 f32 layout: v0 = K0|K2, v1 = K1|K3).
    const float* aF = mem_fringe + (size_t)row * FEATURE_DIM + half * 2;
    const float* aZ = mem_fenzi  + (size_t)row * FEATURE_DIM + half * 2;
    const float* aM = mem_fenmu  + (size_t)row * FEATURE_DIM + half * 2;

    // B-matrix: column n of B comes from one fixed feature vector.
    const bool  valid = (n < 3);
    const float* fsel = (n == 1) ? fea_fenzi : ((n == 2) ? fea_fenmu : fea_f);
    const float* fb   = fsel + b * FEATURE_DIM + half * 2;

    v8f accZ = {};
    v8f accM = {};
    v8f accF = {};

#pragma unroll
    for (int c = 0; c < 32; ++c) {                    // 32 chunks x K=4 = 128
        v2f az = *(const v2f*)(aZ + c * 4);
        v2f am = *(const v2f*)(aM + c * 4);
        v2f af = *(const v2f*)(aF + c * 4);
        v2f bv = *(const v2f*)(fb + c * 4);
        bv.x = valid ? bv.x : 0.0f;                   // zero unused B columns
        bv.y = valid ? bv.y : 0.0f;
        accZ = __builtin_amdgcn_wmma_f32_16x16x4_f32(false, az, false, bv,
                                                     (short)0, accZ, false, false);
        accM = __builtin_amdgcn_wmma_f32_16x16x4_f32(false, am, false, bv,
                                                     (short)0, accM, false, false);
        accF = __builtin_amdgcn_wmma_f32_16x16x4_f32(false, af, false, bv,
                                                     (short)0, accF, false, false);
    }

    // Stage D-matrix columns {0,1,2} into LDS for coalesced global stores.
    // D layout: lane holds column N = lane&15; VGPR i holds M = i + 8*(lane>>4).
    if (n == 0) {
#pragma unroll
        for (int i = 0; i < 8; ++i) {
            sout[w][0 * 16 + i + 8 * half] = accZ[i] * INV_T;   // f_fenzi
            sout[w][1 * 16 + i + 8 * half] = accM[i] * INV_T;   // f_fenmu
        }
    } else if (n == 1) {
#pragma unroll
        for (int i = 0; i < 8; ++i) {
            sout[w][2 * 16 + i + 8 * half] = accF[i] * INV_T;   // fenzi_f
            sout[w][3 * 16 + i + 8 * half] = accM[i] * INV_T;   // fenzi_fenmu
        }
    } else if (n == 2) {
#pragma unroll
        for (int i = 0; i < 8; ++i) {
            sout[w][4 * 16 + i + 8 * half] = accF[i] * INV_T;   // fenmu_f
            sout[w][5 * 16 + i + 8 * half] = accZ[i] * INV_T;   // fenmu_fenzi
        }
    }
    __syncthreads();

#pragma unroll
    for (int p = 0; p < 3; ++p) {
        const int j = p * 2 + half;                   // output family 0..5
        out[(size_t)j * (BATCH * DATA_SIZE) + (size_t)b * DATA_SIZE + k0 + n] =
            sout[w][j * 16 + n];
    }
}

// ---------------------------------------------------------------------------
// Kernel 2: copy the three banks into the output regions (float4, grid-stride)
// ---------------------------------------------------------------------------
__global__ __launch_bounds__(256)
void copy_banks_kernel(const float4* __restrict__ f,
                       const float4* __restrict__ z,
                       const float4* __restrict__ m,
                       float4* __restrict__ out)
{
    const int N4  = DATA_SIZE * FEATURE_DIM / 4;      // 1,048,576 per bank
    const int tot = 3 * N4;
    for (int i = blockIdx.x * blockDim.x + threadIdx.x; i < tot;
         i += gridDim.x * blockDim.x) {
        const int bank = i / N4;
        const int off  = i - bank * N4;
        const float4* src = (bank == 0) ? f : (bank == 1) ? z : m;
        out[(size_t)bank * N4 + off] = src[off];
    }
}

// ---------------------------------------------------------------------------
// Kernel 3: momentum blend + L2 normalize + scatter (48 rows total).
// One wave per (bank, batch) pair; __shfl_xor butterfly for the row norm.
// ---------------------------------------------------------------------------
__global__ __launch_bounds__(32)
void momentum_update_kernel(const float* __restrict__ fea_f,
                            const float* __restrict__ fea_fenzi,
                            const float* __restrict__ fea_fenmu,
                            const int*   __restrict__ y,
                            const float* __restrict__ mem_fringe,
                            const float* __restrict__ mem_fenzi,
                            const float* __restrict__ mem_fenmu,
                            float* __restrict__ out_base)
{
    const int bank = blockIdx.x >> 4;
    const int b    = blockIdx.x & 15;
    const int lane = threadIdx.x;
    const size_t bankN = (size_t)DATA_SIZE * FEATURE_DIM;

    const float* mem;
    const float* fea;
    float* dst;
    if (bank == 0)      { mem = mem_fringe; fea = fea_f;     dst = out_base; }
    else if (bank == 1) { mem = mem_fenzi;  fea = fea_fenzi; dst = out_base + bankN; }
    else                { mem = mem_fenmu;  fea = fea_fenmu; dst = out_base + 2 * bankN; }

    const int row = y[b];
    float4 m = *(const float4*)(mem + (size_t)row * FEATURE_DIM + lane * 4);
    float4 f = *(const float4*)(fea + b * FEATURE_DIM + lane * 4);

    float4 p;
    p.x = (m.x + f.x) * 0.5f;                         // momentum = 0.5
    p.y = (m.y + f.y) * 0.5f;
    p.z = (m.z + f.z) * 0.5f;
    p.w = (m.w + f.w) * 0.5f;

    float ss = p.x * p.x + p.y * p.y + p.z * p.z + p.w * p.w;
#pragma unroll
    for (int off = 16; off >= 1; off >>= 1)
        ss += __shfl_xor(ss, off, 32);

    const float inv = 1.0f / sqrtf(ss);
    p.x *= inv; p.y *= inv; p.z *= inv; p.w *= inv;

    *(float4*)(dst + (size_t)row * FEATURE_DIM + lane * 4) = p;
}

// ---------------------------------------------------------------------------
extern "C" void kernel_launch(void* const* d_in, const int* in_sizes, int n_in,
                              void* d_out, int out_size, void* d_ws, size_t ws_size,
                              hipStream_t stream) {
    const float* fea_f      = (const float*)d_in[0];
    const float* fea_fenzi  = (const float*)d_in[1];
    const float* fea_fenmu  = (const float*)d_in[2];
    const int*   y          = (const int*)d_in[3];
    const int*   idx        = (const int*)d_in[4];
    const float* mem_fringe = (const float*)d_in[5];
    const float* mem_fenzi  = (const float*)d_in[6];
    const float* mem_fenmu  = (const float*)d_in[7];

    float* out = (float*)d_out;
    float* new_banks = out + (size_t)6 * BATCH * DATA_SIZE;   // 16B-aligned

    // 1) six logit families (WMMA): 16*2048 waves, 8 waves/block
    dots_wmma_kernel<<<4096, 256, 0, stream>>>(
        fea_f, fea_fenzi, fea_fenmu, idx,
        mem_fringe, mem_fenzi, mem_fenmu, out);

    // 2) bank copy into output regions
    copy_banks_kernel<<<4096, 256, 0, stream>>>(
        (const float4*)mem_fringe, (const float4*)mem_fenzi,
        (const float4*)mem_fenmu, (float4*)new_banks);

    // 3) scatter the 48 momentum-updated, normalized rows (after the copy)
    momentum_update_kernel<<<48, 32, 0, stream>>>(
        fea_f, fea_fenzi, fea_fenmu, y,
        mem_fringe, mem_fenzi, mem_fenmu, new_banks);
}